// NeuralODE_15616501088263
// MI455X (gfx1250) — compile-verified
//
#include <hip/hip_runtime.h>
#include <stdint.h>

#define BATCH   131072
#define STEPS   499
#define HID     50
#define HHALF   25
#define ROWELTS (BATCH * 2)   // elements per time-row of the output

typedef uint32_t u32x4 __attribute__((ext_vector_type(4)));
typedef int      i32x8 __attribute__((ext_vector_type(8)));
typedef int      i32x4 __attribute__((ext_vector_type(4)));

// CDNA5 hardware tanh (TRANS pipe, co-executes with VALU FMAs).
__device__ __forceinline__ float fast_tanh(float v) {
#if __has_builtin(__builtin_amdgcn_tanhf)
  return __builtin_amdgcn_tanhf(v);
#else
  float r;
  // ISA §7.4: TRANS result needs 1 independent op / V_NOP before use.
  asm volatile("v_tanh_f32 %0, %1\n\tv_nop" : "=v"(r) : "v"(v));
  return r;
#endif
}

// Tensor Data Mover: 1-D copy of n fp32 words global -> LDS.
// D# built per CDNA5 ISA ch.8: group0 {count=1, lds_addr, global_addr, type=2},
// group1 {data_size=4B, tensor_dim0=n, tensor_dim1=1, tile_dim0=n, tile_dim1=1,
// stride0=n}; groups 2/3 zero (tensor <= 2D).
__device__ __forceinline__ void tdm_load_1d(uint32_t lds_off, const void* gptr,
                                            uint32_t n) {
#if __has_builtin(__builtin_amdgcn_tensor_load_to_lds)
  uint64_t ga = (uint64_t)(uintptr_t)gptr;
  u32x4 g0 = {0u, 0u, 0u, 0u};
  g0[0] = 1u;                                             // count=1 (user D#)
  g0[1] = lds_off;                                        // lds_addr (bytes)
  g0[2] = (uint32_t)ga;                                   // global_addr[31:0]
  g0[3] = (uint32_t)((ga >> 32) & 0x01FFFFFFu) | (2u << 30); // addr[56:32]|type=2
  i32x8 g1 = {0, 0, 0, 0, 0, 0, 0, 0};
  g1[0] = (int)(2u << 16);                 // data_size=2 -> 4 bytes/elt
  g1[1] = (int)((n & 0xFFFFu) << 16);      // tensor_dim0[15:0]   (bits 63:48)
  g1[2] = (int)((n >> 16) | (1u << 16));   // tensor_dim0[31:16] | tensor_dim1=1
  g1[3] = (int)((n & 0xFFFFu) << 16);      // tile_dim0 (bits 127:112)
  g1[4] = 1;                               // tile_dim1 = 1
  g1[5] = (int)n;                          // tensor_dim0_stride[31:0]
  i32x4 z4 = {0, 0, 0, 0};
  i32x8 z8 = {0, 0, 0, 0, 0, 0, 0, 0};
  // 6-arg form (clang-23 / therock-10.0 HIP): g0, g1, g2, g3, extra, cpol.
  __builtin_amdgcn_tensor_load_to_lds(g0, g1, z4, z4, z8, 0);
#else
  (void)lds_off; (void)gptr; (void)n;
#endif
}

__global__ __launch_bounds__(256) void ode_euler_kernel(
    const float* __restrict__ y0, const float* __restrict__ tg,
    const float* __restrict__ w1, const float* __restrict__ b1,
    const float* __restrict__ w2, const float* __restrict__ b2,
    float* __restrict__ out) {
  __shared__ float s_w1[100];   // [50][2] row-major
  __shared__ float s_b1[52];
  __shared__ float s_w2[100];   // [2][50] row-major
  __shared__ float s_b2[4];
  __shared__ float s_t[500];

  const int tid = threadIdx.x;
#if __has_builtin(__builtin_amdgcn_tensor_load_to_lds)
  if (tid == 0) {   // TDM is wave-level (EXEC-ignored); issue from one wave
    tdm_load_1d((uint32_t)(uintptr_t)(void*)s_w1, w1, 100);
    tdm_load_1d((uint32_t)(uintptr_t)(void*)s_b1, b1, 50);
    tdm_load_1d((uint32_t)(uintptr_t)(void*)s_w2, w2, 100);
    tdm_load_1d((uint32_t)(uintptr_t)(void*)s_b2, b2, 2);
    tdm_load_1d((uint32_t)(uintptr_t)(void*)s_t,  tg, 500);
    __builtin_amdgcn_s_wait_tensorcnt(0);
  }
#else
  for (int i = tid; i < 100; i += 256) { s_w1[i] = w1[i]; s_w2[i] = w2[i]; }
  for (int i = tid; i < 50;  i += 256) s_b1[i] = b1[i];
  if (tid < 2) s_b2[tid] = b2[tid];
  for (int i = tid; i < 500; i += 256) s_t[i] = tg[i];
#endif
  __syncthreads();

  // Two lanes per batch element: parity p owns hidden units [25p, 25p+25).
  const int p = tid & 1;
  const int e = blockIdx.x * 128 + (tid >> 1);
  const int j0 = p * HHALF;

  float w1x[HHALF], w1y[HHALF], bb[HHALF], w20[HHALF], w21[HHALF];
#pragma unroll
  for (int i = 0; i < HHALF; ++i) {
    const int j = j0 + i;
    w1x[i] = s_w1[2 * j];
    w1y[i] = s_w1[2 * j + 1];
    bb[i]  = s_b1[j];
    w20[i] = s_w2[j];
    w21[i] = s_w2[HID + j];
  }
  const float b2a = p ? 0.0f : s_b2[0];   // bias counted once per pair
  const float b2b = p ? 0.0f : s_b2[1];

  float x = y0[2 * e];
  float y = y0[2 * e + 1];

  // Lane-pair stores adjacent floats -> fully coalesced 128B/wave, NT (524MB
  // output stream > 192MB L2, keep it out of the cache).
  float* op = out + (2 * e + p);
  __builtin_nontemporal_store(p ? y : x, op);

  for (int s = 0; s < STEPS; ++s) {
    const float dt = s_t[s + 1] - s_t[s];
    float pf0 = b2a, pf1 = b2b;
#pragma unroll
    for (int i = 0; i < HHALF; ++i) {
      float pre = __builtin_fmaf(w1y[i], y, __builtin_fmaf(w1x[i], x, bb[i]));
      float h = fast_tanh(pre);
      pf0 = __builtin_fmaf(w20[i], h, pf0);
      pf1 = __builtin_fmaf(w21[i], h, pf1);
    }
    // Butterfly across the lane pair: both lanes end with the full f0/f1.
    const float f0 = pf0 + __shfl_xor(pf0, 1, 32);
    const float f1 = pf1 + __shfl_xor(pf1, 1, 32);
    x = __builtin_fmaf(dt, f0, x);
    y = __builtin_fmaf(dt, f1, y);
    op += ROWELTS;
    __builtin_nontemporal_store(p ? y : x, op);
  }
}

extern "C" void kernel_launch(void* const* d_in, const int* in_sizes, int n_in,
                              void* d_out, int out_size, void* d_ws,
                              size_t ws_size, hipStream_t stream) {
  const float* y0 = (const float*)d_in[0];
  const float* t  = (const float*)d_in[1];
  const float* w1 = (const float*)d_in[2];
  const float* b1 = (const float*)d_in[3];
  const float* w2 = (const float*)d_in[4];
  const float* b2 = (const float*)d_in[5];
  (void)in_sizes; (void)n_in; (void)out_size; (void)d_ws; (void)ws_size;

  dim3 grid(BATCH / 128);   // 128 elements per 256-thread block (2 lanes/elt)
  dim3 block(256);
  ode_euler_kernel<<<grid, block, 0, stream>>>(y0, t, w1, b1, w2, b2,
                                               (float*)d_out);
}